// GCBFGraphNet_90159953477735
// MI455X (gfx1250) — compile-verified
//
#include <hip/hip_runtime.h>

#define HID    128
#define NODE_F 16
#define EDGE_F 16
#define STEPS  3
#define LDA    260   // padded LDS leading dim for A (floats); keeps 16B row alignment, kills bank conflicts

typedef __attribute__((ext_vector_type(2))) float v2f;
typedef __attribute__((ext_vector_type(8))) float v8f;

__device__ __forceinline__ float lrelu(float x) { return x > 0.0f ? x : 0.01f * x; }

// ---------------- zero workspace region ----------------
__global__ void k_zero(float* __restrict__ p, long n) {
  long i = (long)blockIdx.x * blockDim.x + threadIdx.x;
  if (i < n) p[i] = 0.0f;
}

// ---------------- fold edge-encoder + message nets:  Weff_i = We @ Wm[i], beff_i = be @ Wm[i] + bm[i]
__global__ void k_prep(const float* __restrict__ We, const float* __restrict__ be,
                       const float* __restrict__ Wm, const float* __restrict__ bm,
                       float* __restrict__ Weff, float* __restrict__ beff) {
  int i = blockIdx.x;      // step
  int n = threadIdx.x;     // 0..127 output column
  const float* Wmi = Wm + (long)i * HID * HID;
  for (int f = 0; f < EDGE_F; ++f) {
    float s = 0.0f;
    for (int j = 0; j < HID; ++j) s += We[f * HID + j] * Wmi[j * HID + n];
    Weff[((long)i * EDGE_F + f) * HID + n] = s;
  }
  float sb = 0.0f;
  for (int j = 0; j < HID; ++j) sb += be[j] * Wmi[j * HID + n];
  beff[(long)i * HID + n] = sb + bm[(long)i * HID + n];
}

// ---------------- scatter: S[r] += edges[e], cnt[r] += 1  (segment-sum of raw edge features)
__global__ void k_scatter(const float* __restrict__ edges, const int* __restrict__ recv,
                          float* __restrict__ S, float* __restrict__ cnt, int E) {
  int e = blockIdx.x * blockDim.x + threadIdx.x;
  if (e >= E) return;
  int r = recv[e];
  const float4* er = (const float4*)(edges + (long)e * EDGE_F);
  float* sr = S + (long)r * EDGE_F;
#pragma unroll
  for (int q = 0; q < 4; ++q) {
    float4 v = er[q];
    atomicAdd(sr + 4 * q + 0, v.x);
    atomicAdd(sr + 4 * q + 1, v.y);
    atomicAdd(sr + 4 * q + 2, v.z);
    atomicAdd(sr + 4 * q + 3, v.w);
  }
  atomicAdd(cnt + r, 1.0f);
}

// ---------------- node encoder: emb = nodes @ Wn + bn
__global__ void k_encode(const float* __restrict__ nodes, const float* __restrict__ Wn,
                         const float* __restrict__ bn, float* __restrict__ emb, int N) {
  int node = blockIdx.x * 2 + (threadIdx.x / HID);
  int c = threadIdx.x % HID;
  if (node >= N) return;
  const float* nr = nodes + (long)node * NODE_F;
  float s = bn[c];
#pragma unroll
  for (int f = 0; f < NODE_F; ++f) s += nr[f] * Wn[f * HID + c];
  emb[(long)node * HID + c] = s;
}

// ---------------- one message-passing step:
//   agg  = S @ Weff + cnt (.) beff               [N,128]
//   emb' = lrelu([emb | agg] @ Wu + bu)          [N,128]
// Block: 256 threads (8 waves), 32 node rows (2 WMMA M-tiles), each wave owns 16 output cols.
__global__ void __launch_bounds__(256) k_step(
    const float* __restrict__ emb_in, const float* __restrict__ S, const float* __restrict__ cnt,
    const float* __restrict__ Weff, const float* __restrict__ beff,
    const float* __restrict__ Wu, const float* __restrict__ bu,
    float* __restrict__ emb_out, int N) {
  __shared__ float As[32 * LDA];        // A = [emb | agg], 32 rows x 256 cols (padded)
  __shared__ float Ss[32 * 16];         // S tile
  __shared__ float Cs[32];              // degree tile
  __shared__ float Wf[EDGE_F * HID];    // Weff staged
  __shared__ float Bf[HID];             // beff staged

  const int tid = threadIdx.x;
  const long base = (long)blockIdx.x * 32;

  // stage Weff / beff
  for (int t = tid; t < EDGE_F * HID / 4; t += 256)
    ((float4*)Wf)[t] = ((const float4*)Weff)[t];
  if (tid < HID) Bf[tid] = beff[tid];

  // load emb half of A: rows 0..31, cols 0..127
  {
    int r = tid >> 3;
    int c0 = (tid & 7) * 16;
    long g = base + r;
    float4* dst = (float4*)(&As[r * LDA + c0]);
    if (g < N) {
      const float4* src = (const float4*)(emb_in + g * HID + c0);
      dst[0] = src[0]; dst[1] = src[1]; dst[2] = src[2]; dst[3] = src[3];
    } else {
      float4 z = {0.f, 0.f, 0.f, 0.f};
      dst[0] = z; dst[1] = z; dst[2] = z; dst[3] = z;
    }
  }
  // load S tile + degrees
  if (tid < 32) {
    long g = base + tid;
    float4* dst = (float4*)(&Ss[tid * 16]);
    if (g < N) {
      const float4* src = (const float4*)(S + g * EDGE_F);
      dst[0] = src[0]; dst[1] = src[1]; dst[2] = src[2]; dst[3] = src[3];
      Cs[tid] = cnt[g];
    } else {
      float4 z = {0.f, 0.f, 0.f, 0.f};
      dst[0] = z; dst[1] = z; dst[2] = z; dst[3] = z;
      Cs[tid] = 0.0f;
    }
  }
  __syncthreads();

  // agg half of A: As[r][128+c] = sum_j Ss[r][j]*Weff[j][c] + cnt[r]*beff[c]
  {
    int r = tid >> 3;
    int c0 = (tid & 7) * 16;
    float cn = Cs[r];
    const float* srow = &Ss[r * 16];
    for (int c = c0; c < c0 + 16; ++c) {
      float s = cn * Bf[c];
#pragma unroll
      for (int j = 0; j < EDGE_F; ++j) s += srow[j] * Wf[j * HID + c];
      As[r * LDA + HID + c] = s;
    }
  }
  __syncthreads();

  // WMMA: C[16x16] per (wave, M-tile); K = 256 in steps of 4 (V_WMMA_F32_16X16X4_F32)
  const int wave = tid >> 5;
  const int lane = tid & 31;
  const int kh = lane >> 4;        // K-half selector per ISA f32 A/B layout
  const int m  = lane & 15;        // A row / C col index
  const int nc = wave * 16 + (lane & 15);  // output column this lane touches via B

  v8f acc0 = {};
  v8f acc1 = {};
  for (int k = 0; k < 2 * HID; k += 4) {
    int kk = k + 2 * kh;
    v2f a0; a0.x = As[m * LDA + kk];        a0.y = As[m * LDA + kk + 1];
    v2f a1; a1.x = As[(m + 16) * LDA + kk]; a1.y = As[(m + 16) * LDA + kk + 1];
    v2f b;  b.x  = Wu[(long)kk * HID + nc]; b.y  = Wu[(long)(kk + 1) * HID + nc];
    acc0 = __builtin_amdgcn_wmma_f32_16x16x4_f32(false, a0, false, b, (short)0, acc0, false, false);
    acc1 = __builtin_amdgcn_wmma_f32_16x16x4_f32(false, a1, false, b, (short)0, acc1, false, false);
  }

  // bias + leaky_relu + store; C layout: VGPR v -> row v + 8*(lane>>4), col = lane&15 (+wave*16)
  float bias = bu[nc];
#pragma unroll
  for (int v = 0; v < 8; ++v) {
    int r = v + 8 * kh;
    long g0 = base + r;
    if (g0 < N) emb_out[g0 * HID + nc] = lrelu(acc0[v] + bias);
    long g1 = base + 16 + r;
    if (g1 < N) emb_out[g1 * HID + nc] = lrelu(acc1[v] + bias);
  }
}

// ---------------- h = output_net(emb_final[0]);  grad of output_net(node_encoder(row0)) wrt pos
__global__ void k_final(const float* __restrict__ emb_fin, const float* __restrict__ nodes,
                        const float* __restrict__ Wn, const float* __restrict__ bn,
                        const float* __restrict__ Wo1, const float* __restrict__ bo1,
                        const float* __restrict__ Wo2, const float* __restrict__ bo2,
                        const float* __restrict__ Wo3, const float* __restrict__ bo3,
                        float* __restrict__ out) {
  __shared__ float sx[HID], s1[HID], s2[64];
  __shared__ float se[HID], sz1[HID], sg1[HID], sg2[64], sge[HID];
  int t = threadIdx.x;

  // ---- forward h on final embedding of node 0 ----
  sx[t] = emb_fin[t];
  __syncthreads();
  {
    float z = bo1[t];
    for (int j = 0; j < HID; ++j) z += sx[j] * Wo1[j * HID + t];
    s1[t] = lrelu(z);
  }
  __syncthreads();
  if (t < 64) {
    float z = bo2[t];
    for (int j = 0; j < HID; ++j) z += s1[j] * Wo2[j * 64 + t];
    s2[t] = lrelu(z);
  }
  __syncthreads();
  if (t == 0) {
    float y = bo3[0];
    for (int k = 0; k < 64; ++k) y += s2[k] * Wo3[k];
    out[0] = y;
  }

  // ---- gradient path: emb = row0 @ Wn + bn (row0 == nodes[0] since pos = nodes[0,3:6]) ----
  {
    float e = bn[t];
#pragma unroll
    for (int f = 0; f < NODE_F; ++f) e += nodes[f] * Wn[f * HID + t];
    se[t] = e;
  }
  __syncthreads();
  {
    float z = bo1[t];
    for (int j = 0; j < HID; ++j) z += se[j] * Wo1[j * HID + t];
    sz1[t] = z;
    s1[t] = lrelu(z);
  }
  __syncthreads();
  if (t < 64) {
    float z = bo2[t];
    for (int j = 0; j < HID; ++j) z += s1[j] * Wo2[j * 64 + t];
    sg2[t] = Wo3[t] * (z > 0.0f ? 1.0f : 0.01f);   // dL/dz2
  }
  __syncthreads();
  {
    float g = 0.0f;
    for (int k = 0; k < 64; ++k) g += Wo2[t * 64 + k] * sg2[k];
    sg1[t] = g * (sz1[t] > 0.0f ? 1.0f : 0.01f);   // dL/dz1
  }
  __syncthreads();
  {
    float g = 0.0f;
    for (int kk = 0; kk < HID; ++kk) g += Wo1[t * HID + kk] * sg1[kk];
    sge[t] = g;                                    // dL/demb
  }
  __syncthreads();
  if (t < 3) {
    float g = 0.0f;
    for (int j = 0; j < HID; ++j) g += Wn[(3 + t) * HID + j] * sge[j];
    out[1 + t] = g;
  }
}

extern "C" void kernel_launch(void* const* d_in, const int* in_sizes, int n_in,
                              void* d_out, int out_size, void* d_ws, size_t ws_size,
                              hipStream_t stream) {
  (void)n_in; (void)out_size; (void)ws_size;
  const float* nodes = (const float*)d_in[0];
  const float* edges = (const float*)d_in[1];
  const int*   recv  = (const int*)d_in[2];
  const float* Wn  = (const float*)d_in[3];
  const float* bn  = (const float*)d_in[4];
  const float* We  = (const float*)d_in[5];
  const float* be  = (const float*)d_in[6];
  const float* Wm  = (const float*)d_in[7];
  const float* bm  = (const float*)d_in[8];
  const float* Wu  = (const float*)d_in[9];
  const float* bu  = (const float*)d_in[10];
  // d_in[11]=Wa, d_in[12]=ba: dead (softmax over a size-1 axis is exactly 1)
  const float* Wo1 = (const float*)d_in[13];
  const float* bo1 = (const float*)d_in[14];
  const float* Wo2 = (const float*)d_in[15];
  const float* bo2 = (const float*)d_in[16];
  const float* Wo3 = (const float*)d_in[17];
  const float* bo3 = (const float*)d_in[18];

  const int N = in_sizes[0] / NODE_F;  // 50000
  const int E = in_sizes[2];           // 800000

  float* ws   = (float*)d_ws;
  float* S    = ws;                          // N*16
  float* cnt  = S    + (size_t)N * EDGE_F;   // N
  float* embA = cnt  + (size_t)N;            // N*128
  float* embB = embA + (size_t)N * HID;      // N*128
  float* Weff = embB + (size_t)N * HID;      // 3*16*128
  float* beff = Weff + (size_t)STEPS * EDGE_F * HID;  // 3*128

  long nz = (long)N * EDGE_F + N;
  k_zero<<<(int)((nz + 255) / 256), 256, 0, stream>>>(S, nz);
  k_prep<<<STEPS, HID, 0, stream>>>(We, be, Wm, bm, Weff, beff);
  k_scatter<<<(E + 255) / 256, 256, 0, stream>>>(edges, recv, S, cnt, E);
  k_encode<<<(N + 1) / 2, 256, 0, stream>>>(nodes, Wn, bn, embA, N);

  int sblocks = (N + 31) / 32;
  k_step<<<sblocks, 256, 0, stream>>>(embA, S, cnt, Weff + 0 * EDGE_F * HID, beff + 0 * HID,
                                      Wu + 0L * 2 * HID * HID, bu + 0 * HID, embB, N);
  k_step<<<sblocks, 256, 0, stream>>>(embB, S, cnt, Weff + 1 * EDGE_F * HID, beff + 1 * HID,
                                      Wu + 1L * 2 * HID * HID, bu + 1 * HID, embA, N);
  k_step<<<sblocks, 256, 0, stream>>>(embA, S, cnt, Weff + 2 * EDGE_F * HID, beff + 2 * HID,
                                      Wu + 2L * 2 * HID * HID, bu + 2 * HID, embB, N);

  k_final<<<1, HID, 0, stream>>>(embB, nodes, Wn, bn, Wo1, bo1, Wo2, bo2, Wo3, bo3, (float*)d_out);
}